// AdaFastFoodMergedModel_50568944943731
// MI455X (gfx1250) — compile-verified
//
#include <hip/hip_runtime.h>
#include <math.h>

// ---------------------------------------------------------------------------
// FastFood transform: y = SCALE * (H G Pi H B x)[row_idx]
//   x: 8192 x 4096 f32, FWHT length 4096 = 16^3
// Strategy: each workgroup (256 thr = 8 wave32) processes 4 rows.
//   - TDM async-loads the 4x4096 f32 tile into LDS (64 KB), s_wait_tensorcnt
//   - FWHT4096 = H16 applied along each base-16 digit (strides 1,16,256),
//     each H16 tile done as 4 chained V_WMMA_F32_16X16X4_F32 (exact fp32)
//   - permutation+G into second LDS buffer, second FWHT, subsampled store
// Memory-bound: ~174 MB total traffic -> ~7.5us floor @ 23.3 TB/s.
// ---------------------------------------------------------------------------

typedef __attribute__((ext_vector_type(2))) float        v2f;
typedef __attribute__((ext_vector_type(8))) float        v8f;
typedef __attribute__((ext_vector_type(4))) unsigned int u32x4;
typedef __attribute__((ext_vector_type(4))) int          i32x4;
typedef __attribute__((ext_vector_type(8))) int          i32x8;

#define NROWS    8192
#define DLEN     4096
#define MOUT     1228
#define RPW      4            // rows per workgroup
#define NTHREADS 256
#define NWAVES   8
#define NVEC     (RPW * (DLEN / 16))   // 16-vectors per digit pass = 1024

// Sylvester Hadamard entry: H[m][k] = (-1)^popcount(m & k), exact in fp32.
__device__ __forceinline__ float hsign(int m, int k) {
    return (__builtin_popcount(m & k) & 1) ? -1.0f : 1.0f;
}

// One digit-axis pass of the radix-16 FWHT over the whole LDS tile.
// vshift in {0,4,8}: element stride = 1<<vshift, batch base mapping
//   base(v) = ((v >> vshift) << (vshift+4)) | (v & (stride-1))
// Each 16x16 tile: D = H16 * X via 4 chained f32 WMMAs; scaled by 1/4
// (3 passes -> 1/64 = 1/sqrt(4096), the orthonormal FWHT scale).
// `wave` is an SGPR (readfirstlane'd) so the tile loop is scalar-branched
// and EXEC stays statically all-ones around the WMMAs.
__device__ __forceinline__ void fwht16_pass(float* __restrict__ buf,
                                            int vshift, int lane, int wave) {
    const int stride = 1 << vshift;
    const int nhi    = lane >> 4;     // which half-wave
    const int nlo    = lane & 15;     // N column (and M row for A operand)
    const int ntiles = NVEC / 16;     // 64
    for (int t = wave; t < ntiles; t += NWAVES) {
        const int v    = t * 16 + nlo;
        const int base = ((v >> vshift) << (vshift + 4)) | (v & (stride - 1));
        v8f acc = {};
#pragma unroll
        for (int kc = 0; kc < 4; ++kc) {
            // B operand (4x16 slice of X): VGPR0 rows kc*4+{0,1}, VGPR1 rows kc*4+{2,3}
            v2f bop, aop;
            bop.x = buf[base + (kc * 4 + nhi) * stride];
            bop.y = buf[base + (kc * 4 + 2 + nhi) * stride];
            // A operand (16x4 slice of H16): VGPR0 K=kc*4+{0|2}, VGPR1 K+1 (lane-half split)
            const int ka = kc * 4 + (nhi ? 2 : 0);
            aop.x = hsign(nlo, ka);
            aop.y = hsign(nlo, ka + 1);
            acc = __builtin_amdgcn_wmma_f32_16x16x4_f32(
                false, aop, false, bop, (short)0, acc, false, false);
        }
        // D layout: lanes 0-15 N=lane rows 0..7, lanes 16-31 N=lane-16 rows 8..15
#pragma unroll
        for (int i = 0; i < 8; ++i) {
            const int m = i + nhi * 8;
            buf[base + m * stride] = acc[i] * 0.25f;
        }
    }
}

__global__ void __launch_bounds__(NTHREADS)
fastfood_fwht_kernel(const float* __restrict__ x,
                     const float* __restrict__ Bsign,
                     const float* __restrict__ G,
                     const int*   __restrict__ Pi,
                     const int*   __restrict__ ridx,
                     float*       __restrict__ out) {
    extern __shared__ float smem[];
    float* bufA = smem;               // 4 x 4096 f32 = 64 KB
    float* bufB = smem + RPW * DLEN;  // second 64 KB (permutation target)

    const int tid  = threadIdx.x;
    const int lane = tid & 31;
    // Provably wave-uniform wave id -> scalar loops, EXEC stays full for WMMA.
    const int wave = __builtin_amdgcn_readfirstlane(tid >> 5);
    const int row0 = blockIdx.x * RPW;

    // ---- Stage 0: bring the 4x4096 f32 tile into LDS via the TDM ----------
#if __has_builtin(__builtin_amdgcn_tensor_load_to_lds)
    if (wave == 0) {
        const unsigned long long gaddr =
            (unsigned long long)(uintptr_t)(x + (size_t)row0 * DLEN);
        const unsigned int ldsoff = (unsigned int)(uintptr_t)bufA;
        u32x4 g0;
        g0[0] = 1u;                                        // count=1, user D#
        g0[1] = ldsoff;                                    // lds_addr
        g0[2] = (unsigned int)(gaddr & 0xFFFFFFFFull);     // global_addr[31:0]
        g0[3] = (unsigned int)((gaddr >> 32) & 0x1FFFFFFull) | 0x80000000u; // [56:32] + type=2
        i32x8 g1;
        g1[0] = (int)(2u << 16);                           // data_size = 4 bytes
        g1[1] = (int)((DLEN & 0xFFFF) << 16);              // tensor_dim0[15:0]
        g1[2] = (int)((DLEN >> 16) | ((NROWS & 0xFFFF) << 16)); // dim0 hi | dim1 lo
        g1[3] = (int)((NROWS >> 16) | ((DLEN & 0xFFFF) << 16)); // dim1 hi | tile_dim0
        g1[4] = (int)RPW;                                  // tile_dim1 (| tile_dim2=0)
        g1[5] = (int)DLEN;                                 // tensor_dim0_stride[31:0]
        g1[6] = 0;                                         // stride0 hi | stride1 lo
        g1[7] = 0;
        i32x4 gz = {0, 0, 0, 0};
#if __has_include(<hip/amd_detail/amd_gfx1250_TDM.h>)
        i32x8 gz8 = {0, 0, 0, 0, 0, 0, 0, 0};
        __builtin_amdgcn_tensor_load_to_lds(g0, g1, gz, gz, gz8, 0);
#else
        __builtin_amdgcn_tensor_load_to_lds(g0, g1, gz, gz, 0);
#endif
        __builtin_amdgcn_s_wait_tensorcnt(0);
    }
#else
    for (int i = tid; i < RPW * DLEN; i += NTHREADS)
        bufA[i] = x[(size_t)row0 * DLEN + i];
#endif
    __syncthreads();

    // ---- Stage 1: sign flip (h = x * B), vectorized b128 ------------------
    {
        float4*       a4 = (float4*)bufA;
        const float4* b4 = (const float4*)Bsign;
        for (int i = tid; i < RPW * DLEN / 4; i += NTHREADS) {
            const float4 s = b4[i & (DLEN / 4 - 1)];
            float4 v = a4[i];
            v.x *= s.x; v.y *= s.y; v.z *= s.z; v.w *= s.w;
            a4[i] = v;
        }
    }
    __syncthreads();

    // ---- Stage 2: first orthonormal FWHT-4096 (3 radix-16 WMMA passes) ----
    fwht16_pass(bufA, 0, lane, wave); __syncthreads();
    fwht16_pass(bufA, 4, lane, wave); __syncthreads();
    fwht16_pass(bufA, 8, lane, wave); __syncthreads();

    // ---- Stage 3: permutation gather + Gaussian diag: B[j] = A[Pi[j]]*G[j] -
    for (int i = tid; i < RPW * DLEN; i += NTHREADS) {
        const int r = i >> 12;
        const int j = i & (DLEN - 1);
        bufB[(r << 12) + j] = bufA[(r << 12) + Pi[j]] * G[j];
    }
    __syncthreads();

    // ---- Stage 4: second orthonormal FWHT-4096 ----------------------------
    fwht16_pass(bufB, 0, lane, wave); __syncthreads();
    fwht16_pass(bufB, 4, lane, wave); __syncthreads();
    fwht16_pass(bufB, 8, lane, wave); __syncthreads();

    // ---- Stage 5: subsample + scale, stream to HBM ------------------------
    const float scale = sqrtf((float)DLEN / (float)MOUT);  // sqrt(L/M)
    for (int i = tid; i < RPW * MOUT; i += NTHREADS) {
        const int r = i / MOUT;
        const int j = i - r * MOUT;
        out[(size_t)(row0 + r) * MOUT + j] = bufB[(r << 12) + ridx[j]] * scale;
    }
}

extern "C" void kernel_launch(void* const* d_in, const int* in_sizes, int n_in,
                              void* d_out, int out_size, void* d_ws, size_t ws_size,
                              hipStream_t stream) {
    const float* x     = (const float*)d_in[0];
    const float* Bsign = (const float*)d_in[1];
    const float* G     = (const float*)d_in[2];
    const int*   Pi    = (const int*)d_in[3];
    const int*   ridx  = (const int*)d_in[4];
    float*       out   = (float*)d_out;

    const dim3 grid(NROWS / RPW);     // 2048 workgroups
    const dim3 block(NTHREADS);       // 8 wave32 per block
    const size_t smem = (size_t)2 * RPW * DLEN * sizeof(float);  // 128 KB

    fastfood_fwht_kernel<<<grid, block, smem, stream>>>(x, Bsign, G, Pi, ridx, out);
}